// DFsmn_21586505629750
// MI455X (gfx1250) — compile-verified
//
#include <hip/hip_runtime.h>
#include <stdint.h>

// DFSMN: p[t,d] = (1+l0[d])*v[t,d] + sum_{k=1..10} r[k-1,d]*v[t+k,d]
//                 + sum_{k=1..19} l[k,d]*p[t-k,d]
// One thread per (b,d) chain. Register-resident IIR state; HBM stream staged
// through LDS with the gfx1250 async global->LDS engine (ASYNCcnt) when the
// toolchain exposes it, else direct nontemporal loads.

#if defined(__has_builtin)
#if __has_builtin(__builtin_amdgcn_global_load_async_to_lds_b32) && \
    __has_builtin(__builtin_amdgcn_s_wait_asynccnt)
#define USE_ASYNC_LDS 1
#endif
#endif
#ifndef USE_ASYNC_LDS
#define USE_ASYNC_LDS 0
#endif

namespace {

constexpr int TDIM = 2048;
constexpr int DDIM = 512;
constexpr int BDIM = 32;
constexpr int KL   = 19;        // recurrent taps l_filter[1..19]
constexpr int KR   = 10;        // lookahead (FIR) taps
constexpr int U    = 8;         // time steps per chunk
constexpr int W    = KR + U;    // register sliding window of v (18 floats)
constexpr int PF   = 64;        // L2 prefetch distance in time rows
constexpr int NTHR = 256;

// Direct path: loads idx = t0+18..t0+25 must be < TDIM  -> t0 <= 2022
constexpr int T_MAIN_DIRECT = ((TDIM - W - U) / U) * U + U;          // 2024
// Async path: issues idx = t0+26..t0+33 must be < TDIM  -> t0 <= 2014
constexpr int T_MAIN_ASYNC  = ((TDIM - W - 2 * U) / U) * U + U;      // 2016

#if USE_ASYNC_LDS
// Signature probed from compiler diagnostics:
//   __builtin_amdgcn_global_load_async_to_lds_b32(AS1 int* gsrc, AS3 int* ldst,
//                                                 imm offset, imm cpol)
typedef __attribute__((address_space(1))) int gint_as1;   // global source
typedef __attribute__((address_space(3))) int lint_as3;   // LDS destination
#endif

// ---- compute core: U outputs from the current register window ------------
__device__ __forceinline__ void dfsmn_steps(
    int t0, float* __restrict__ ob, float l0,
    const float (&lc)[KL], const float (&rc)[KR],
    const float (&vwin)[W], float (&ph)[KL])
{
#pragma unroll
  for (int u = 0; u < U; ++u) {
    float s[4];                       // 4-way split: dep chain ~8 not 30
    s[0] = l0 * vwin[u];
    s[1] = 0.0f; s[2] = 0.0f; s[3] = 0.0f;
#pragma unroll
    for (int k = 1; k <= KR; ++k)
      s[k & 3] = fmaf(rc[k - 1], vwin[u + k], s[k & 3]);
#pragma unroll
    for (int k = 0; k < KL; ++k)
      s[(KR + 1 + k) & 3] = fmaf(lc[k], ph[k], s[(KR + 1 + k) & 3]);
    const float p = (s[0] + s[1]) + (s[2] + s[3]);

    __builtin_nontemporal_store(p, ob + (size_t)(t0 + u) * DDIM);

#pragma unroll
    for (int k = KL - 1; k > 0; --k) ph[k] = ph[k - 1];   // SSA renaming
    ph[0] = p;
  }
}

// ---- tail chunk: guarded direct loads with zero padding past T -----------
__device__ __forceinline__ void chunk_guarded(
    int t0, const float* __restrict__ vb, float* __restrict__ ob,
    float l0, const float (&lc)[KL], const float (&rc)[KR],
    float (&vwin)[W], float (&ph)[KL])
{
  float inc[U];
#pragma unroll
  for (int u = 0; u < U; ++u) {
    const int idx = t0 + W + u;              // wave-uniform
    float x = 0.0f;
    if (idx < TDIM) x = __builtin_nontemporal_load(vb + (size_t)idx * DDIM);
    inc[u] = x;
  }
  dfsmn_steps(t0, ob, l0, lc, rc, vwin, ph);
#pragma unroll
  for (int i = 0; i < W - U; ++i) vwin[i] = vwin[i + U];
#pragma unroll
  for (int u = 0; u < U; ++u) vwin[W - U + u] = inc[u];
}

#if !USE_ASYNC_LDS
// ---- fallback main chunk: direct unguarded loads -------------------------
__device__ __forceinline__ void chunk_direct(
    int t0, const float* __restrict__ vb, float* __restrict__ ob,
    float l0, const float (&lc)[KL], const float (&rc)[KR],
    float (&vwin)[W], float (&ph)[KL])
{
  if (t0 + PF < TDIM)
    __builtin_prefetch(vb + (size_t)(t0 + PF) * DDIM, 0, 0);
  float inc[U];
#pragma unroll
  for (int u = 0; u < U; ++u)
    inc[u] = __builtin_nontemporal_load(vb + (size_t)(t0 + W + u) * DDIM);
  dfsmn_steps(t0, ob, l0, lc, rc, vwin, ph);
#pragma unroll
  for (int i = 0; i < W - U; ++i) vwin[i] = vwin[i + U];
#pragma unroll
  for (int u = 0; u < U; ++u) vwin[W - U + u] = inc[u];
}
#endif

__global__ __launch_bounds__(NTHR) void dfsmn_iir_kernel(
    const float* __restrict__ v, const float* __restrict__ lfil,
    const float* __restrict__ rfil, float* __restrict__ out)
{
  const int tid = threadIdx.x;
  const int d = blockIdx.x * NTHR + tid;     // lanes = consecutive d
  const int b = blockIdx.y;
  const float* vb = v   + (size_t)b * TDIM * DDIM + d;
  float*       ob = out + (size_t)b * TDIM * DDIM + d;

  // Per-channel coefficients -> registers (coalesced loads).
  const float l0 = 1.0f + lfil[d];
  float lc[KL];
#pragma unroll
  for (int k = 0; k < KL; ++k) lc[k] = lfil[(k + 1) * DDIM + d];
  float rc[KR];
#pragma unroll
  for (int k = 0; k < KR; ++k) rc[k] = rfil[k * DDIM + d];

  float ph[KL];
#pragma unroll
  for (int k = 0; k < KL; ++k) ph[k] = 0.0f;

  float vwin[W];                             // v[0 .. W-1]
#pragma unroll
  for (int i = 0; i < W; ++i)
    vwin[i] = __builtin_nontemporal_load(vb + (size_t)i * DDIM);

  int t0 = 0;

#if USE_ASYNC_LDS
  // Double-buffered LDS staging; each thread owns 8 contiguous floats per
  // stage -> splice reads are two ds_load_b128, no barriers needed (private
  // slots + same-wave ASYNCcnt ordering).
  __shared__ float stage[2][NTHR][U];        // 16 KB

  // Prologue: buf0 <- v[18..25] (always in-bounds).
#pragma unroll
  for (int u = 0; u < U; ++u)
    __builtin_amdgcn_global_load_async_to_lds_b32(
        (gint_as1*)(vb + (size_t)(W + u) * DDIM),
        (lint_as3*)&stage[0][tid][u], 0, 0);

  int s = 0;
  for (; t0 < T_MAIN_ASYNC; t0 += U) {
    if (t0 + PF < TDIM)
      __builtin_prefetch(vb + (size_t)(t0 + PF) * DDIM, 0, 0);

    // Issue next chunk's tail: v[t0+26 .. t0+33] -> back buffer.
    const int ns = s ^ 1;
#pragma unroll
    for (int u = 0; u < U; ++u)
      __builtin_amdgcn_global_load_async_to_lds_b32(
          (gint_as1*)(vb + (size_t)(t0 + W + U + u) * DDIM),
          (lint_as3*)&stage[ns][tid][u], 0, 0);

    // ~240 FMAs of latency cover before we need the front buffer.
    dfsmn_steps(t0, ob, l0, lc, rc, vwin, ph);

    // Allow the 8 just-issued to remain in flight; front buffer is done.
    __builtin_amdgcn_s_wait_asynccnt(U);

#pragma unroll
    for (int i = 0; i < W - U; ++i) vwin[i] = vwin[i + U];
    const float4 a = *reinterpret_cast<const float4*>(&stage[s][tid][0]);
    const float4 c = *reinterpret_cast<const float4*>(&stage[s][tid][4]);
    vwin[W - U + 0] = a.x; vwin[W - U + 1] = a.y;
    vwin[W - U + 2] = a.z; vwin[W - U + 3] = a.w;
    vwin[W - U + 4] = c.x; vwin[W - U + 5] = c.y;
    vwin[W - U + 6] = c.z; vwin[W - U + 7] = c.w;
    s = ns;
  }
#else
  for (; t0 < T_MAIN_DIRECT; t0 += U)
    chunk_direct(t0, vb, ob, l0, lc, rc, vwin, ph);
#endif

  // Tail: guarded direct loads, zero padding past T.
  for (; t0 < TDIM; t0 += U)
    chunk_guarded(t0, vb, ob, l0, lc, rc, vwin, ph);
}

} // namespace

extern "C" void kernel_launch(void* const* d_in, const int* in_sizes, int n_in,
                              void* d_out, int out_size, void* d_ws, size_t ws_size,
                              hipStream_t stream) {
  (void)in_sizes; (void)n_in; (void)out_size; (void)d_ws; (void)ws_size;
  const float* v  = (const float*)d_in[0];   // (32,1,2048,512) fp32
  const float* lf = (const float*)d_in[1];   // (20,512) fp32
  const float* rf = (const float*)d_in[2];   // (10,512) fp32
  float* out = (float*)d_out;                // (32,1,2048,512) fp32

  dim3 grid(DDIM / NTHR, BDIM, 1);           // 64 blocks x 256 threads
  dfsmn_iir_kernel<<<grid, dim3(NTHR, 1, 1), 0, stream>>>(v, lf, rf, out);
}